// cross_deformable_conv_59863254172257
// MI455X (gfx1250) — compile-verified
//
#include <hip/hip_runtime.h>
#include <hip/hip_bf16.h>

// ---------------------------------------------------------------------------
// MI455X (gfx1250) multi-scale deformable attention block.
// All GEMM-shaped work runs on V_WMMA_F32_16X16X4_F32 (exact fp32 math).
// One wave computes a 16(M) x 64(N) tile: 4 f32 accumulators (32 VGPRs),
// A fragment loaded once per K-step and reused across 4 WMMAs to cut the
// VMEM-issue rate (the bottleneck of the naive 16x16 version) by ~2x.
// 128-thread blocks = 4 waves, each owning a different M-tile.
// ---------------------------------------------------------------------------

typedef __attribute__((ext_vector_type(2))) float v2f;
typedef __attribute__((ext_vector_type(8))) float v8f;

#define N_BATCH 2
#define C_DIM   256
#define HEADS   8
#define CHD     32
#define LEVELS  4
#define SAMPLES 4
#define LN_EPS  1e-5f

__device__ __forceinline__ v8f wmma4(v2f a, v2f b, v8f c) {
  // (neg_a, A, neg_b, B, c_mod, C, reuse_a, reuse_b)
  return __builtin_amdgcn_wmma_f32_16x16x4_f32(false, a, false, b, (short)0, c,
                                               false, false);
}

// ---------------------------------------------------------------------------
// Y[n][co][p] = sum_ci W[co][ci] * X[n][ci][p] + bias[co] (+ R[n][co][p])
// Wave tile: 16 co x 64 pixels. grid = (HW/64, Cout/64, N), block = 128.
// A (16x4): lane m = lane&15, k = k0 + 2*(lane>>4) + {0,1} (b64 load).
// B (4x16): lane n = lane&15, same k split.
// C/D: vgpr r -> M = r + 8*(lane>>4), N = lane&15.
// ---------------------------------------------------------------------------
__global__ void conv1x1_wmma(const float* __restrict__ X,
                             const float* __restrict__ W,
                             const float* __restrict__ bias,
                             const float* __restrict__ R,
                             float* __restrict__ Y,
                             int Cin, int Cout, int HW) {
  const int lane = threadIdx.x & 31;
  const int wv   = threadIdx.x >> 5;
  const int half = lane >> 4;
  const int lm   = lane & 15;
  const int p0   = blockIdx.x * 64;
  const int co0  = blockIdx.y * 64 + wv * 16;
  const int n    = blockIdx.z;

  const float* Xn = X + (size_t)n * Cin * HW + p0 + lm;
  const float* Wa = W + (size_t)(co0 + lm) * Cin;

  v8f c0 = {}, c1 = {}, c2 = {}, c3 = {};
  for (int k = 0; k < Cin; k += 4) {
    const int kk = k + 2 * half;
    v2f a;
    a.x = Wa[kk];
    a.y = Wa[kk + 1];
    const float* xr0 = Xn + (size_t)kk * HW;
    const float* xr1 = Xn + (size_t)(kk + 1) * HW;
    v2f b0 = {xr0[0],  xr1[0]};
    v2f b1 = {xr0[16], xr1[16]};
    v2f b2 = {xr0[32], xr1[32]};
    v2f b3 = {xr0[48], xr1[48]};
    c0 = wmma4(a, b0, c0);
    c1 = wmma4(a, b1, c1);
    c2 = wmma4(a, b2, c2);
    c3 = wmma4(a, b3, c3);
  }
  const size_t base = (size_t)n * Cout * HW;
  for (int r = 0; r < 8; ++r) {
    const int m = co0 + r + 8 * half;
    const float bm = bias[m];
    const size_t row = base + (size_t)m * HW + p0 + lm;
    float v0 = c0[r] + bm, v1 = c1[r] + bm, v2 = c2[r] + bm, v3 = c3[r] + bm;
    if (R) { v0 += R[row]; v1 += R[row + 16]; v2 += R[row + 32]; v3 += R[row + 48]; }
    Y[row]      = v0;
    Y[row + 16] = v1;
    Y[row + 32] = v2;
    Y[row + 48] = v3;
  }
}

// ---------------------------------------------------------------------------
// Implicit-GEMM 3x3 conv, pad=1, stride=1. K = 9*Cin, taps outer so the
// zero-pad predicates are K-loop-invariant. Wave tile: 16 co x 64 pixels.
// ---------------------------------------------------------------------------
__global__ void conv3x3_wmma(const float* __restrict__ Q,
                             const float* __restrict__ W,
                             const float* __restrict__ bias,
                             float* __restrict__ Y,
                             int Cin, int Cout, int H, int Wd) {
  const int HW   = H * Wd;
  const int lane = threadIdx.x & 31;
  const int wv   = threadIdx.x >> 5;
  const int half = lane >> 4;
  const int lm   = lane & 15;
  const int p0   = blockIdx.x * 64;
  const int co0  = blockIdx.y * 64 + wv * 16;
  const int n    = blockIdx.z;

  int py[4], px[4];
  for (int t = 0; t < 4; ++t) {
    const int p = p0 + 16 * t + lm;
    py[t] = p / Wd;
    px[t] = p % Wd;
  }

  const float* Qn = Q + (size_t)n * Cin * HW;
  const float* Wc = W + (size_t)(co0 + lm) * Cin * 9;

  v8f c0 = {}, c1 = {}, c2 = {}, c3 = {};
  for (int tap = 0; tap < 9; ++tap) {
    const int ky = tap / 3 - 1;
    const int kx = tap % 3 - 1;
    bool ok[4];
    const float* qb[4];
    for (int t = 0; t < 4; ++t) {
      const int yy = py[t] + ky;
      const int xx = px[t] + kx;
      ok[t] = (yy >= 0) && (yy < H) && (xx >= 0) && (xx < Wd);
      qb[t] = Qn + (size_t)yy * Wd + xx;
    }
    const float* wb = Wc + tap;
    for (int k = 0; k < Cin; k += 4) {
      const int kk = k + 2 * half;
      v2f a;
      a.x = wb[(size_t)kk * 9];
      a.y = wb[(size_t)(kk + 1) * 9];
      const size_t o0 = (size_t)kk * HW;
      const size_t o1 = (size_t)(kk + 1) * HW;
      v2f b0 = {ok[0] ? qb[0][o0] : 0.f, ok[0] ? qb[0][o1] : 0.f};
      v2f b1 = {ok[1] ? qb[1][o0] : 0.f, ok[1] ? qb[1][o1] : 0.f};
      v2f b2 = {ok[2] ? qb[2][o0] : 0.f, ok[2] ? qb[2][o1] : 0.f};
      v2f b3 = {ok[3] ? qb[3][o0] : 0.f, ok[3] ? qb[3][o1] : 0.f};
      c0 = wmma4(a, b0, c0);
      c1 = wmma4(a, b1, c1);
      c2 = wmma4(a, b2, c2);
      c3 = wmma4(a, b3, c3);
    }
  }
  const size_t base = (size_t)n * Cout * HW;
  for (int r = 0; r < 8; ++r) {
    const int m = co0 + r + 8 * half;
    const float bm = bias[m];
    const size_t row = base + (size_t)m * HW + p0 + lm;
    Y[row]      = c0[r] + bm;
    Y[row + 16] = c1[r] + bm;
    Y[row + 32] = c2[r] + bm;
    Y[row + 48] = c3[r] + bm;
  }
}

// ---------------------------------------------------------------------------
// Row-major GEMM: Y[m][nf] = act(sum_k X[m][k]*W[k][nf] + bias[nf]) (+Res)
// Wave tile: 16 m x 64 nf. grid = (Nf/64, M/64), block = 128.
// A rows are K-contiguous (b64 loads); B rows N-contiguous per K.
// ---------------------------------------------------------------------------
__global__ void ffn_gemm_wmma(const float* __restrict__ X,
                              const float* __restrict__ W,
                              const float* __restrict__ bias,
                              const float* __restrict__ Res,
                              float* __restrict__ Y,
                              int K, int Nf, int relu) {
  const int lane = threadIdx.x & 31;
  const int wv   = threadIdx.x >> 5;
  const int half = lane >> 4;
  const int lm   = lane & 15;
  const int n0   = blockIdx.x * 64;
  const int m0   = blockIdx.y * 64 + wv * 16;

  const float* Xa = X + (size_t)(m0 + lm) * K;
  v8f c0 = {}, c1 = {}, c2 = {}, c3 = {};
  for (int k = 0; k < K; k += 4) {
    const int kk = k + 2 * half;
    v2f a;
    a.x = Xa[kk];
    a.y = Xa[kk + 1];
    const float* wr0 = W + (size_t)kk * Nf + n0 + lm;
    const float* wr1 = W + (size_t)(kk + 1) * Nf + n0 + lm;
    v2f b0 = {wr0[0],  wr1[0]};
    v2f b1 = {wr0[16], wr1[16]};
    v2f b2 = {wr0[32], wr1[32]};
    v2f b3 = {wr0[48], wr1[48]};
    c0 = wmma4(a, b0, c0);
    c1 = wmma4(a, b1, c1);
    c2 = wmma4(a, b2, c2);
    c3 = wmma4(a, b3, c3);
  }
  for (int r = 0; r < 8; ++r) {
    const int m = m0 + r + 8 * half;
    const size_t row = (size_t)m * Nf + n0 + lm;
    float v0 = c0[r] + bias[n0 + lm];
    float v1 = c1[r] + bias[n0 + 16 + lm];
    float v2 = c2[r] + bias[n0 + 32 + lm];
    float v3 = c3[r] + bias[n0 + 48 + lm];
    if (relu) {
      v0 = fmaxf(v0, 0.f); v1 = fmaxf(v1, 0.f);
      v2 = fmaxf(v2, 0.f); v3 = fmaxf(v3, 0.f);
    }
    if (Res) { v0 += Res[row]; v1 += Res[row + 16]; v2 += Res[row + 32]; v3 += Res[row + 48]; }
    Y[row]      = v0;
    Y[row + 16] = v1;
    Y[row + 32] = v2;
    Y[row + 48] = v3;
  }
}

// q = feat + pos + level_embed[c]
__global__ void make_q(const float* __restrict__ f, const float* __restrict__ pos,
                       const float* __restrict__ le, float* __restrict__ q,
                       int HW, int total) {
  const int i = blockIdx.x * blockDim.x + threadIdx.x;
  if (i >= total) return;
  const int c = (i / HW) % C_DIM;
  q[i] = f[i] + pos[i] + le[c];
}

// In-place softmax over the 16 taps of wgt[n][h*16+tap][p].
__global__ void softmax16(float* __restrict__ wgt, int HW) {
  const int t = blockIdx.x * blockDim.x + threadIdx.x;
  const int total = N_BATCH * HEADS * HW;
  if (t >= total) return;
  const int p  = t % HW;
  const int nh = t / HW;
  const int n  = nh / HEADS;
  const int h  = nh % HEADS;
  float* base = wgt + ((size_t)n * (HEADS * 16) + h * 16) * HW + p;
  float v[16];
  float mx = -3.402823466e38f;
  for (int i = 0; i < 16; ++i) { v[i] = base[(size_t)i * HW]; mx = fmaxf(mx, v[i]); }
  float s = 0.f;
  for (int i = 0; i < 16; ++i) { v[i] = expf(v[i] - mx); s += v[i]; }
  const float inv = 1.f / s;
  for (int i = 0; i < 16; ++i) base[(size_t)i * HW] = v[i] * inv;
}

// ---------------------------------------------------------------------------
// Deformable bilinear sampling + weighted aggregation.
// One wave per (n, head, pixel); lane = channel within head (CHD == 32).
// grid_sample semantics: bilinear, zeros padding, align_corners=False.
// ---------------------------------------------------------------------------
__global__ void deform_sample(const float* __restrict__ values,
                              const float* __restrict__ off,
                              const float* __restrict__ wgt,
                              const float* __restrict__ points,
                              float* __restrict__ agg,
                              int H, int Wd, float inv_stride) {
  const int HW   = H * Wd;
  const int wid  = (int)((blockIdx.x * blockDim.x + threadIdx.x) >> 5);
  const int lane = threadIdx.x & 31;
  const int total = N_BATCH * HEADS * HW;
  if (wid >= total) return;
  const int p  = wid % HW;
  const int nh = wid / HW;
  const int n  = nh / HEADS;
  const int h  = nh % HEADS;

  const float px = points[p * 2 + 0] * inv_stride;  // == x + 0.5 in query units
  const float py = points[p * 2 + 1] * inv_stride;  // == y + 0.5

  const int    sizes[LEVELS] = {64, 32, 16, 8};
  const size_t voff[LEVELS]  = {0, 2097152, 2621440, 2752512};

  float acc = 0.f;
  for (int j = 0; j < LEVELS; ++j) {
    const int sj = sizes[j];
    const float* val =
        values + voff[j] + ((size_t)n * C_DIM + h * CHD + lane) * (size_t)(sj * sj);
    for (int s = 0; s < SAMPLES; ++s) {
      const int cb = h * 32 + j * 8 + s * 2;  // offset channels (h, l, s, d)
      const float ox = off[((size_t)n * C_DIM + cb + 0) * HW + p];
      const float oy = off[((size_t)n * C_DIM + cb + 1) * HW + p];
      const float w  = wgt[((size_t)n * (HEADS * 16) + h * 16 + j * 4 + s) * HW + p];
      const float lx = (ox + px) / (float)Wd * 2.f - 1.f;
      const float ly = (oy + py) / (float)H  * 2.f - 1.f;
      const float xf = (lx + 1.f) * ((float)sj * 0.5f) - 0.5f;
      const float yf = (ly + 1.f) * ((float)sj * 0.5f) - 0.5f;
      const float x0f = floorf(xf), y0f = floorf(yf);
      const int   x0 = (int)x0f,   y0 = (int)y0f;
      const float wx1 = xf - x0f, wx0 = 1.f - wx1;
      const float wy1 = yf - y0f, wy0 = 1.f - wy1;
      const int x1 = x0 + 1, y1 = y0 + 1;
      const bool bx0 = (x0 >= 0) && (x0 < sj);
      const bool bx1 = (x1 >= 0) && (x1 < sj);
      const bool by0 = (y0 >= 0) && (y0 < sj);
      const bool by1 = (y1 >= 0) && (y1 < sj);
      const float v00 = (bx0 && by0) ? val[(size_t)y0 * sj + x0] : 0.f;
      const float v10 = (bx1 && by0) ? val[(size_t)y0 * sj + x1] : 0.f;
      const float v01 = (bx0 && by1) ? val[(size_t)y1 * sj + x0] : 0.f;
      const float v11 = (bx1 && by1) ? val[(size_t)y1 * sj + x1] : 0.f;
      acc += w * (v00 * wx0 * wy0 + v10 * wx1 * wy0 +
                  v01 * wx0 * wy1 + v11 * wx1 * wy1);
    }
  }
  agg[((size_t)n * C_DIM + h * CHD + lane) * HW + p] = acc;
}

// ---------------------------------------------------------------------------
// LayerNorm over C=256 per pixel. One wave per pixel, 8 channels per lane,
// shfl_xor butterfly reduction.
// ---------------------------------------------------------------------------
__global__ void ln_nchw_to_rows(const float* __restrict__ X,
                                const float* __restrict__ g,
                                const float* __restrict__ b,
                                float* __restrict__ Y, int HW) {
  const int wid  = (int)((blockIdx.x * blockDim.x + threadIdx.x) >> 5);
  const int lane = threadIdx.x & 31;
  const int total = N_BATCH * HW;
  if (wid >= total) return;
  const int n = wid / HW;
  const int p = wid % HW;
  const float* Xn = X + (size_t)n * C_DIM * HW + p;
  float v[8], s = 0.f, s2 = 0.f;
  for (int r = 0; r < 8; ++r) {
    v[r] = Xn[(size_t)(lane + 32 * r) * HW];
    s += v[r]; s2 += v[r] * v[r];
  }
  for (int o = 16; o >= 1; o >>= 1) {
    s  += __shfl_xor(s,  o, 32);
    s2 += __shfl_xor(s2, o, 32);
  }
  const float mu  = s * (1.f / C_DIM);
  const float var = s2 * (1.f / C_DIM) - mu * mu;
  const float inv = rsqrtf(var + LN_EPS);
  float* Yr = Y + (size_t)wid * C_DIM;
  for (int r = 0; r < 8; ++r) {
    const int c = lane + 32 * r;
    Yr[c] = (v[r] - mu) * inv * g[c] + b[c];
  }
}

__global__ void ln_rows_to_nchw(const float* __restrict__ X,
                                const float* __restrict__ g,
                                const float* __restrict__ b,
                                float* __restrict__ Y, int HW) {
  const int wid  = (int)((blockIdx.x * blockDim.x + threadIdx.x) >> 5);
  const int lane = threadIdx.x & 31;
  const int total = N_BATCH * HW;
  if (wid >= total) return;
  const int n = wid / HW;
  const int p = wid % HW;
  const float* Xr = X + (size_t)wid * C_DIM;
  float v[8], s = 0.f, s2 = 0.f;
  for (int r = 0; r < 8; ++r) {
    v[r] = Xr[lane + 32 * r];
    s += v[r]; s2 += v[r] * v[r];
  }
  for (int o = 16; o >= 1; o >>= 1) {
    s  += __shfl_xor(s,  o, 32);
    s2 += __shfl_xor(s2, o, 32);
  }
  const float mu  = s * (1.f / C_DIM);
  const float var = s2 * (1.f / C_DIM) - mu * mu;
  const float inv = rsqrtf(var + LN_EPS);
  float* Yn = Y + (size_t)n * C_DIM * HW + p;
  for (int r = 0; r < 8; ++r) {
    const int c = lane + 32 * r;
    Yn[(size_t)c * HW] = (v[r] - mu) * inv * g[c] + b[c];
  }
}

// ---------------------------------------------------------------------------
// Host-side launch sequence
// ---------------------------------------------------------------------------
extern "C" void kernel_launch(void* const* d_in, const int* in_sizes, int n_in,
                              void* d_out, int out_size, void* d_ws, size_t ws_size,
                              hipStream_t stream) {
  (void)in_sizes; (void)n_in; (void)out_size;

  static const int    SZ[LEVELS]   = {64, 32, 16, 8};
  static const int    STRD[LEVELS] = {8, 16, 32, 64};
  static const size_t LOFF[LEVELS] = {0, 2097152, 2621440, 2752512};  // floats

  const float* feat[LEVELS];
  const float* pos[LEVELS];
  const float* pts[LEVELS];
  for (int i = 0; i < LEVELS; ++i) {
    feat[i] = (const float*)d_in[3 * i + 0];
    pos[i]  = (const float*)d_in[3 * i + 1];
    pts[i]  = (const float*)d_in[3 * i + 2];
  }
  const float* Wv   = (const float*)d_in[12];
  const float* bv   = (const float*)d_in[13];
  const float* Woff = (const float*)d_in[14];
  const float* boff = (const float*)d_in[15];
  const float* Ww   = (const float*)d_in[16];
  const float* bw   = (const float*)d_in[17];
  const float* le   = (const float*)d_in[18];
  const float* Wn   = (const float*)d_in[19];
  const float* bn   = (const float*)d_in[20];
  const float* g1   = (const float*)d_in[21];
  const float* b1   = (const float*)d_in[22];
  const float* W1   = (const float*)d_in[23];
  const float* bf1  = (const float*)d_in[24];
  const float* W2   = (const float*)d_in[25];
  const float* bf2  = (const float*)d_in[26];
  const float* g2   = (const float*)d_in[27];
  const float* b2   = (const float*)d_in[28];

  const size_t NEED = 2785280ull + 2097152ull * 6 + 1048576ull + 4194304ull;
  if (ws_size < NEED * sizeof(float)) return;
  float* ws     = (float*)d_ws;
  float* values = ws;                   // 2,785,280 (all levels, persistent)
  float* qbuf   = values + 2785280;
  float* offb   = qbuf   + 2097152;
  float* wgtb   = offb   + 2097152;
  float* aggb   = wgtb   + 1048576;
  float* outb   = aggb   + 2097152;
  float* xln    = outb   + 2097152;
  float* hb     = xln    + 2097152;
  float* yb     = hb     + 4194304;

  float* out = (float*)d_out;

  // Stage 1: values[j] = 1x1 conv (Wv) of each level's features.
  for (int j = 0; j < LEVELS; ++j) {
    const int HW = SZ[j] * SZ[j];
    conv1x1_wmma<<<dim3(HW / 64, C_DIM / 64, N_BATCH), 128, 0, stream>>>(
        feat[j], Wv, bv, nullptr, values + LOFF[j], C_DIM, C_DIM, HW);
  }

  // Stage 2: per-level deformable attention + FFN.
  for (int i = 0; i < LEVELS; ++i) {
    const int S  = SZ[i];
    const int HW = S * S;
    const int totq = N_BATCH * C_DIM * HW;

    make_q<<<(totq + 255) / 256, 256, 0, stream>>>(feat[i], pos[i],
                                                   le + i * C_DIM, qbuf, HW, totq);

    conv3x3_wmma<<<dim3(HW / 64, C_DIM / 64, N_BATCH), 128, 0, stream>>>(
        qbuf, Woff, boff, offb, C_DIM, C_DIM, S, S);
    conv3x3_wmma<<<dim3(HW / 64, (HEADS * 16) / 64, N_BATCH), 128, 0, stream>>>(
        qbuf, Ww, bw, wgtb, C_DIM, HEADS * 16, S, S);

    const int totw = N_BATCH * HEADS * HW;
    softmax16<<<(totw + 255) / 256, 256, 0, stream>>>(wgtb, HW);

    const int sampThreads = N_BATCH * HEADS * HW * 32;  // one wave per (n,h,p)
    deform_sample<<<(sampThreads + 255) / 256, 256, 0, stream>>>(
        values, offb, wgtb, pts[i], aggb, S, S, 1.f / (float)STRD[i]);

    conv1x1_wmma<<<dim3(HW / 64, C_DIM / 64, N_BATCH), 128, 0, stream>>>(
        aggb, Wn, bn, feat[i], outb, C_DIM, C_DIM, HW);

    const int lnThreads = N_BATCH * HW * 32;
    ln_nchw_to_rows<<<(lnThreads + 255) / 256, 256, 0, stream>>>(outb, g1, b1,
                                                                 xln, HW);

    const int Mrows = N_BATCH * HW;
    ffn_gemm_wmma<<<dim3((2 * C_DIM) / 64, Mrows / 64), 128, 0, stream>>>(
        xln, W1, bf1, nullptr, hb, C_DIM, 2 * C_DIM, 1);
    ffn_gemm_wmma<<<dim3(C_DIM / 64, Mrows / 64), 128, 0, stream>>>(
        hb, W2, bf2, xln, yb, 2 * C_DIM, C_DIM, 0);

    ln_rows_to_nchw<<<(lnThreads + 255) / 256, 256, 0, stream>>>(yb, g2, b2,
                                                                 out + LOFF[i], HW);
  }
}